// VariationalQuantumLayer_35605278883844
// MI455X (gfx1250) — compile-verified
//
#include <hip/hip_runtime.h>
#include <math.h>
#include <stdint.h>

typedef float v2f __attribute__((ext_vector_type(2)));
typedef float v8f __attribute__((ext_vector_type(8)));

#define NQ    8
#define DIM   256            // 2^NQ amplitudes
#define NL    3
#define MBLK  32             // batch rows per workgroup (2 M-tiles)
#define AROW  260            // A LDS row stride (floats); 260 % 64 == 4 -> conflict-free
#define KC    32             // K-chunk staged per async step
#define NC    (DIM / KC)     // 8 chunks
#define BROW  36             // B LDS row stride (floats); 36 % 64 spacing-4 pattern, 144B (16B mult)
#define BWAVE (2 * 64 * BROW)        // floats per wave (double buffer, 64 rows each)
#define SMEM_FLOATS (MBLK * AROW + 8 * BWAVE)

#define WMMA(acc, a, b) \
  acc = __builtin_amdgcn_wmma_f32_16x16x4_f32(false, (a), false, (b), (short)0, (acc), false, false)

// ---------------------------------------------------------------------------
// Kernel 1: build circuit unitary U (DIM x DIM complex); weights shared across
// batch. Block j evolves |j>; thread t owns amplitude t. Row-major storage:
// Ure[n*DIM + k] = Re U[n,k];  Uim immediately after (so re rows 0..255 and
// im rows 256..511 form one contiguous 512x256 matrix "Ucat").
// Wire w lives at bit (NQ-1-w) (MSB-first convention).
// ---------------------------------------------------------------------------
__global__ __launch_bounds__(DIM) void build_unitary(const float* __restrict__ w,
                                                     float* __restrict__ Ure,
                                                     float* __restrict__ Uim) {
  __shared__ float re[DIM];
  __shared__ float im[DIM];
  const int t   = threadIdx.x;
  const int col = blockIdx.x;
  re[t] = (t == col) ? 1.f : 0.f;
  im[t] = 0.f;
  __syncthreads();

  for (int l = 0; l < NL; ++l) {
    for (int c = 0; c < NQ; ++c) {          // ring of CNOTs
      const int tg = (c + 1) % NQ;
      const int cm = 1 << (NQ - 1 - c);
      const int tm = 1 << (NQ - 1 - tg);
      if ((t & cm) && !(t & tm)) {
        const int p = t | tm;
        float tr = re[t], ti = im[t];
        re[t] = re[p]; im[t] = im[p];
        re[p] = tr;    im[p] = ti;
      }
      __syncthreads();
    }
    for (int q = 0; q < NQ; ++q) {          // RX, RY, RZ per qubit
      const int   m   = 1 << (NQ - 1 - q);
      const float thx = 0.5f * w[(l * NQ + q) * 3 + 0];
      const float thy = 0.5f * w[(l * NQ + q) * 3 + 1];
      const float thz = 0.5f * w[(l * NQ + q) * 3 + 2];
      {  // RX: [[c, -i s], [-i s, c]]
        const float cg = cosf(thx), sg = sinf(thx);
        if (!(t & m)) {
          const int i1 = t | m;
          float a0r = re[t], a0i = im[t], a1r = re[i1], a1i = im[i1];
          re[t]  = cg * a0r + sg * a1i;
          im[t]  = cg * a0i - sg * a1r;
          re[i1] = cg * a1r + sg * a0i;
          im[i1] = cg * a1i - sg * a0r;
        }
        __syncthreads();
      }
      {  // RY: [[c, -s], [s, c]]
        const float cg = cosf(thy), sg = sinf(thy);
        if (!(t & m)) {
          const int i1 = t | m;
          float a0r = re[t], a0i = im[t], a1r = re[i1], a1i = im[i1];
          re[t]  = cg * a0r - sg * a1r;
          im[t]  = cg * a0i - sg * a1i;
          re[i1] = sg * a0r + cg * a1r;
          im[i1] = sg * a0i + cg * a1i;
        }
        __syncthreads();
      }
      {  // RZ: diag(e^{-i th/2}, e^{+i th/2})
        const float ang = (t & m) ? thz : -thz;
        const float cz = cosf(ang), sz = sinf(ang);
        const float r0 = re[t], i0 = im[t];
        re[t] = r0 * cz - i0 * sz;
        im[t] = r0 * sz + i0 * cz;
        __syncthreads();
      }
    }
  }
  Ure[t * DIM + col] = re[t];
  Uim[t * DIM + col] = im[t];
}

// ---------------------------------------------------------------------------
// Kernel 2: fused product-state A build + f32 WMMA GEMM (K=256, re+im) with
// per-wave async B staging into LDS (double-buffered, GLOBAL_LOAD_ASYNC_TO_LDS
// + s_wait_asynccnt; no barriers needed since each wave consumes only its own
// slice), then |psi|^2 + signed Z-expectation epilogue.
// Block = 32 batch rows, 8 waves; wave owns N-tiles {2w, 2w+1} x {re,im} x 2 M-tiles.
// ---------------------------------------------------------------------------
__global__ void __launch_bounds__(256) vq_forward(const float* __restrict__ x,
                                                  const float* __restrict__ Ucat, // 512x256: re rows 0..255, im rows 256..511
                                                  float* __restrict__ out) {
  extern __shared__ float smem[];
  float* sA = smem;                       // [MBLK][AROW]
  float* sB = smem + MBLK * AROW;         // [8 waves][2 bufs][64 rows][BROW]

  const int tid   = threadIdx.x;
  const int lane  = tid & 31;
  const int wave  = tid >> 5;
  const int mbase = blockIdx.x * MBLK;
  const int nt0   = wave * 2, nt1 = nt0 + 1;
  const int nc    = lane & 15;            // column within 16-wide N tile
  const int kL    = (lane >> 4) << 1;     // lane's K sub-offset (0 or 2)

  float* sBw = sB + wave * BWAVE;

  // ---- async-stage one K-chunk of this wave's B slice (64 rows x KC floats) ----
  // row layout in buffer: [0..15]=re nt0, [16..31]=re nt1, [32..47]=im nt0, [48..63]=im nt1
  const int seg  = lane & 7;              // 16B segment within a 32-float row
  const int row0 = lane >> 3;             // 0..3
  auto stage = [&](int buf, int kcbase) {
    float* lbase = sBw + buf * (64 * BROW);
#pragma unroll
    for (int j = 0; j < 16; ++j) {
      const int row = row0 + j * 4;                       // 0..63
      const int q   = row >> 4;
      const int rr  = row & 15;
      const int grow = ((q & 1) ? nt1 : nt0) * 16 + rr + ((q & 2) ? DIM : 0);
      const float* gp = Ucat + grow * DIM + kcbase + seg * 4;
      const uint32_t lp = (uint32_t)(uintptr_t)(lbase + row * BROW + seg * 4);
      asm volatile("global_load_async_to_lds_b128 %0, %1, off"
                   :: "v"(lp), "v"(gp) : "memory");
    }
  };

  stage(0, 0);   // prefetch chunk 0 while we build the A tile

  // ---- build A tile: real product-state amplitudes for MBLK batch rows ----
  {
    const int m = tid >> 3;               // batch row within tile (0..31)
    const int g = tid & 7;                // fills idx = g*32 .. g*32+31
    float cv[NQ], sv[NQ];
#pragma unroll
    for (int i = 0; i < NQ; ++i) {
      const float h = tanhf(x[(mbase + m) * NQ + i]) * 1.5707963267948966f;
      cv[i] = cosf(h);
      sv[i] = sinf(h);
    }
#pragma unroll
    for (int qq = 0; qq < 32; ++qq) {
      const int idx = g * 32 + qq;
      float v = 1.f;
#pragma unroll
      for (int i = 0; i < NQ; ++i)
        v *= ((idx >> (NQ - 1 - i)) & 1) ? sv[i] : cv[i];
      sA[m * AROW + idx] = v;
    }
  }
  __syncthreads();

  // ---- K loop: double-buffered chunks, 8 WMMAs per K-step (2 mt x 2 nt x re/im) ----
  v8f aR00 = {}, aI00 = {}, aR01 = {}, aI01 = {};
  v8f aR10 = {}, aI10 = {}, aR11 = {}, aI11 = {};

  for (int c = 0; c < NC; ++c) {
    if (c + 1 < NC) {
      stage((c + 1) & 1, (c + 1) * KC);
      asm volatile("s_wait_asynccnt %0" :: "n"(16) : "memory");  // chunk c finished (in-order)
    } else {
      asm volatile("s_wait_asynccnt %0" :: "n"(0) : "memory");
    }
    const float* ap = sA + nc * AROW + c * KC + kL;
    const float* bq = sBw + (c & 1) * (64 * BROW) + nc * BROW + kL;
#pragma unroll 2
    for (int kk = 0; kk < KC; kk += 4) {
      const v2f a0  = *(const v2f*)(ap + kk);
      const v2f a1  = *(const v2f*)(ap + kk + 16 * AROW);
      const v2f br0 = *(const v2f*)(bq + kk);
      const v2f br1 = *(const v2f*)(bq + kk + 16 * BROW);
      const v2f bi0 = *(const v2f*)(bq + kk + 32 * BROW);
      const v2f bi1 = *(const v2f*)(bq + kk + 48 * BROW);
      WMMA(aR00, a0, br0); WMMA(aI00, a0, bi0);
      WMMA(aR01, a0, br1); WMMA(aI01, a0, bi1);
      WMMA(aR10, a1, br0); WMMA(aI10, a1, bi0);
      WMMA(aR11, a1, br1); WMMA(aI11, a1, bi1);
    }
  }
  __syncthreads();   // all A reads done -> safe to reuse sA for probs

  // ---- probs into LDS (C layout: VGPR r -> M = r + 8*(lane>>4), N = lane&15) ----
  {
    const int mh = (lane >> 4) << 3;
#pragma unroll
    for (int r = 0; r < 8; ++r) {
      const int m0 = r + mh, m1 = 16 + r + mh;
      sA[m0 * AROW + nt0 * 16 + nc] = aR00[r] * aR00[r] + aI00[r] * aI00[r];
      sA[m0 * AROW + nt1 * 16 + nc] = aR01[r] * aR01[r] + aI01[r] * aI01[r];
      sA[m1 * AROW + nt0 * 16 + nc] = aR10[r] * aR10[r] + aI10[r] * aI10[r];
      sA[m1 * AROW + nt1 * 16 + nc] = aR11[r] * aR11[r] + aI11[r] * aI11[r];
    }
  }
  __syncthreads();

  // ---- Z expectations: out[b, i] = sum_n (+/-1)*prob[b, n] ----
  {
    const int m = tid >> 3;               // 0..31
    const int i = tid & 7;
    const int shift = NQ - 1 - i;
    float acc = 0.f;
    for (int n = 0; n < DIM; ++n) {
      const float p = sA[m * AROW + n];
      acc += ((n >> shift) & 1) ? -p : p;
    }
    out[(mbase + m) * NQ + i] = acc;
  }
}

// ---------------------------------------------------------------------------
extern "C" void kernel_launch(void* const* d_in, const int* in_sizes, int n_in,
                              void* d_out, int out_size, void* d_ws, size_t ws_size,
                              hipStream_t stream) {
  const float* x = (const float*)d_in[0];   // [B, 8]
  const float* w = (const float*)d_in[1];   // [3, 8, 3]
  float* out = (float*)d_out;               // [B, 8]
  float* Ure = (float*)d_ws;                // 256*256 f32
  float* Uim = Ure + DIM * DIM;             // 256*256 f32 (contiguous -> Ucat 512x256)

  const int B = in_sizes[0] / NQ;           // 16384

  build_unitary<<<DIM, DIM, 0, stream>>>(w, Ure, Uim);
  vq_forward<<<B / MBLK, 256, SMEM_FLOATS * sizeof(float), stream>>>(x, Ure, out);
}